// BipartiteGConv_85383949844813
// MI455X (gfx1250) — compile-verified
//
#include <hip/hip_runtime.h>

typedef float v2f __attribute__((ext_vector_type(2)));
typedef float v8f __attribute__((ext_vector_type(8)));

__device__ __forceinline__ v8f wmma_f32(v2f a, v2f b, v8f c) {
  // D = A(16x4, f32) * B(4x16, f32) + C(16x16, f32)
  return __builtin_amdgcn_wmma_f32_16x16x4_f32(false, a, false, b, (short)0, c,
                                               false, false);
}

// Fire-and-forget f32 atomic add (no return -> STOREcnt), device scope.
// Guarantees global_atomic_add_f32 (no CAS-loop fallback).
__device__ __forceinline__ void atomic_fadd_f32(float* p, float v) {
  asm volatile("global_atomic_add_f32 %0, %1, off scope:SCOPE_DEV"
               :
               : "v"((unsigned long long)(uintptr_t)p), "v"(v)
               : "memory");
}

// ---------------------------------------------------------------------------
// GEMM: C[M,64] = A[M,64] @ B[64,64] (+ bias). One wave -> 16x64 output tile.
// B staged in LDS pre-swizzled as K-pair fragments:
//   Bs2[k2*64+n] = {B[2k2][n], B[2k2+1][n]}
// ---------------------------------------------------------------------------
__launch_bounds__(256)
__global__ void gemm64_kernel(const float* __restrict__ A,
                              const float* __restrict__ B,
                              const float* __restrict__ bias,
                              float* __restrict__ C, int M) {
  __shared__ float2 Bs2[32 * 64];  // 16 KB
  for (int i = threadIdx.x; i < 2048; i += 256) {
    int k2 = i >> 6, n = i & 63;
    Bs2[i] = float2{B[(2 * k2) * 64 + n], B[(2 * k2 + 1) * 64 + n]};
  }
  __syncthreads();

  const int wave = threadIdx.x >> 5;
  const int lane = threadIdx.x & 31;
  const int r = lane & 15;   // row-in-tile / col-in-tile
  const int h = lane >> 4;   // lane half

  const int row0 = (blockIdx.x * 8 + wave) * 16;
  if (row0 >= M) return;  // wave-uniform; no barriers after this point

  int rowA = row0 + r;
  if (rowA >= M) rowA = M - 1;  // clamp: keep EXEC all-ones for WMMA
  const float* __restrict__ Arow = A + (size_t)rowA * 64;

  v8f acc0 = {}, acc1 = {}, acc2 = {}, acc3 = {};
#pragma unroll
  for (int k0 = 0; k0 < 64; k0 += 4) {
    const int ka = k0 + 2 * h;
    float2 av = *reinterpret_cast<const float2*>(Arow + ka);
    v2f a; a.x = av.x; a.y = av.y;
    const int k2 = (k0 >> 1) + h;
    const float2* __restrict__ Bk = &Bs2[k2 * 64 + r];
    float2 f0 = Bk[0], f1 = Bk[16], f2 = Bk[32], f3 = Bk[48];
    v2f b0, b1, b2, b3;
    b0.x = f0.x; b0.y = f0.y;  b1.x = f1.x; b1.y = f1.y;
    b2.x = f2.x; b2.y = f2.y;  b3.x = f3.x; b3.y = f3.y;
    acc0 = wmma_f32(a, b0, acc0);
    acc1 = wmma_f32(a, b1, acc1);
    acc2 = wmma_f32(a, b2, acc2);
    acc3 = wmma_f32(a, b3, acc3);
  }

  float bia0 = 0.f, bia1 = 0.f, bia2 = 0.f, bia3 = 0.f;
  if (bias) { bia0 = bias[r]; bia1 = bias[16 + r]; bia2 = bias[32 + r]; bia3 = bias[48 + r]; }

#pragma unroll
  for (int v = 0; v < 8; ++v) {
    const int row = row0 + v + 8 * h;
    if (row < M) {
      float* __restrict__ Crow = C + (size_t)row * 64;
      Crow[r]      = acc0[v] + bia0;
      Crow[16 + r] = acc1[v] + bia1;
      Crow[32 + r] = acc2[v] + bia2;
      Crow[48 + r] = acc3[v] + bia3;
    }
  }
}

// ---------------------------------------------------------------------------
// Concat GEMM: C[M,64] = [A0|A1][M,128] @ B[128,64] + bias
// ---------------------------------------------------------------------------
__launch_bounds__(256)
__global__ void gemm_concat128_kernel(const float* __restrict__ A0,
                                      const float* __restrict__ A1,
                                      const float* __restrict__ B,
                                      const float* __restrict__ bias,
                                      float* __restrict__ C, int M) {
  __shared__ float2 Bs2[64 * 64];  // 32 KB
  for (int i = threadIdx.x; i < 4096; i += 256) {
    int k2 = i >> 6, n = i & 63;
    Bs2[i] = float2{B[(2 * k2) * 64 + n], B[(2 * k2 + 1) * 64 + n]};
  }
  __syncthreads();

  const int wave = threadIdx.x >> 5;
  const int lane = threadIdx.x & 31;
  const int r = lane & 15;
  const int h = lane >> 4;

  const int row0 = (blockIdx.x * 8 + wave) * 16;
  if (row0 >= M) return;

  int rowA = row0 + r;
  if (rowA >= M) rowA = M - 1;
  const float* __restrict__ Arow0 = A0 + (size_t)rowA * 64;
  const float* __restrict__ Arow1 = A1 + (size_t)rowA * 64;

  v8f acc0 = {}, acc1 = {}, acc2 = {}, acc3 = {};
#pragma unroll
  for (int k0 = 0; k0 < 128; k0 += 4) {
    const int ka = k0 + 2 * h;
    const float* src = (k0 < 64) ? (Arow0 + ka) : (Arow1 + (ka - 64));
    float2 av = *reinterpret_cast<const float2*>(src);
    v2f a; a.x = av.x; a.y = av.y;
    const int k2 = (k0 >> 1) + h;
    const float2* __restrict__ Bk = &Bs2[k2 * 64 + r];
    float2 f0 = Bk[0], f1 = Bk[16], f2 = Bk[32], f3 = Bk[48];
    v2f b0, b1, b2, b3;
    b0.x = f0.x; b0.y = f0.y;  b1.x = f1.x; b1.y = f1.y;
    b2.x = f2.x; b2.y = f2.y;  b3.x = f3.x; b3.y = f3.y;
    acc0 = wmma_f32(a, b0, acc0);
    acc1 = wmma_f32(a, b1, acc1);
    acc2 = wmma_f32(a, b2, acc2);
    acc3 = wmma_f32(a, b3, acc3);
  }

  const float bia0 = bias[r], bia1 = bias[16 + r], bia2 = bias[32 + r], bia3 = bias[48 + r];
#pragma unroll
  for (int v = 0; v < 8; ++v) {
    const int row = row0 + v + 8 * h;
    if (row < M) {
      float* __restrict__ Crow = C + (size_t)row * 64;
      Crow[r]      = acc0[v] + bia0;
      Crow[16 + r] = acc1[v] + bia1;
      Crow[32 + r] = acc2[v] + bia2;
      Crow[48 + r] = acc3[v] + bia3;
    }
  }
}

// ---------------------------------------------------------------------------
// Fused edge kernel: per 16-edge tile
//   msg = leaky_relu(rhs[rj] + lhs[lj] + w*We) ; acc = msg @ Wf + bf
//   hardware-atomic scatter-add into outAcc[rj]  (tables are L2-resident)
// ---------------------------------------------------------------------------
__launch_bounds__(256)
__global__ void edge_kernel(const float* __restrict__ rhs,
                            const float* __restrict__ lhs,
                            const int* __restrict__ rj,
                            const int* __restrict__ lj,
                            const float* __restrict__ weights,
                            const float* __restrict__ We,
                            const float* __restrict__ Wf,
                            const float* __restrict__ bf,
                            float* __restrict__ outAcc, int E) {
  __shared__ float2 WfS2[32 * 64];      // 16 KB, K-pair swizzled Wf
  __shared__ float WeS[64];
  __shared__ float bfS[64];
  __shared__ float msgS[8][16 * 64];    // 4 KB per wave (32 KB)
  __shared__ int rjS[8][16];

  for (int i = threadIdx.x; i < 2048; i += 256) {
    int k2 = i >> 6, n = i & 63;
    WfS2[i] = float2{Wf[(2 * k2) * 64 + n], Wf[(2 * k2 + 1) * 64 + n]};
  }
  if (threadIdx.x < 64) { WeS[threadIdx.x] = We[threadIdx.x]; bfS[threadIdx.x] = bf[threadIdx.x]; }
  __syncthreads();

  const int wave = threadIdx.x >> 5;
  const int lane = threadIdx.x & 31;
  const int r = lane & 15;
  const int h = lane >> 4;

  const int eBase = (blockIdx.x * 8 + wave) * 16;

  // ---- gather + edge embed + leaky-relu into per-wave LDS tile ----
  int e = eBase + r;
  if (e >= E) e = E - 1;                // clamp (redundant work, no stores)
  const int idxR = rj[e];
  const int idxL = lj[e];
  const float w = weights[e];
  if (h == 0) rjS[wave][r] = idxR;

  const float4* __restrict__ pr = reinterpret_cast<const float4*>(rhs + (size_t)idxR * 64) + h * 8;
  const float4* __restrict__ pl = reinterpret_cast<const float4*>(lhs + (size_t)idxL * 64) + h * 8;
  const float4* __restrict__ pe = reinterpret_cast<const float4*>(WeS) + h * 8;
  float4* __restrict__ pm = reinterpret_cast<float4*>(&msgS[wave][r * 64]) + h * 8;
#pragma unroll
  for (int i = 0; i < 8; ++i) {
    float4 a = pr[i], b = pl[i], c = pe[i];
    float4 m;
    m.x = a.x + b.x + w * c.x;
    m.y = a.y + b.y + w * c.y;
    m.z = a.z + b.z + w * c.z;
    m.w = a.w + b.w + w * c.w;
    m.x = m.x > 0.f ? m.x : 0.01f * m.x;
    m.y = m.y > 0.f ? m.y : 0.01f * m.y;
    m.z = m.z > 0.f ? m.z : 0.01f * m.z;
    m.w = m.w > 0.f ? m.w : 0.01f * m.w;
    pm[i] = m;
  }
  // per-wave LDS tile is wave-synchronous: DS ops from one wave are in-order.

  // ---- msg(16x64) @ Wf(64x64) via 64x v_wmma_f32_16x16x4_f32 ----
  v8f acc0 = {}, acc1 = {}, acc2 = {}, acc3 = {};
  const float* __restrict__ mrow = &msgS[wave][r * 64];
#pragma unroll
  for (int k0 = 0; k0 < 64; k0 += 4) {
    const int ka = k0 + 2 * h;
    float2 av = *reinterpret_cast<const float2*>(mrow + ka);
    v2f a; a.x = av.x; a.y = av.y;
    const int k2 = (k0 >> 1) + h;
    const float2* __restrict__ Bk = &WfS2[k2 * 64 + r];
    float2 f0 = Bk[0], f1 = Bk[16], f2 = Bk[32], f3 = Bk[48];
    v2f b0, b1, b2, b3;
    b0.x = f0.x; b0.y = f0.y;  b1.x = f1.x; b1.y = f1.y;
    b2.x = f2.x; b2.y = f2.y;  b3.x = f3.x; b3.y = f3.y;
    acc0 = wmma_f32(a, b0, acc0);
    acc1 = wmma_f32(a, b1, acc1);
    acc2 = wmma_f32(a, b2, acc2);
    acc3 = wmma_f32(a, b3, acc3);
  }

  // ---- + bf, then segment scatter-add (hardware f32 atomics, no return) ----
  const float bia0 = bfS[r], bia1 = bfS[16 + r], bia2 = bfS[32 + r], bia3 = bfS[48 + r];
#pragma unroll
  for (int v = 0; v < 8; ++v) {
    const int m = v + 8 * h;
    if (eBase + m < E) {
      const int seg = rjS[wave][m];
      float* __restrict__ orow = outAcc + (size_t)seg * 64;
      atomic_fadd_f32(orow + r,      acc0[v] + bia0);
      atomic_fadd_f32(orow + 16 + r, acc1[v] + bia1);
      atomic_fadd_f32(orow + 32 + r, acc2[v] + bia2);
      atomic_fadd_f32(orow + 48 + r, acc3[v] + bia3);
    }
  }
}

__global__ void zero4_kernel(float4* __restrict__ p, int n4) {
  int i = blockIdx.x * blockDim.x + threadIdx.x;
  if (i < n4) p[i] = float4{0.f, 0.f, 0.f, 0.f};
}

// ---------------------------------------------------------------------------
extern "C" void kernel_launch(void* const* d_in, const int* in_sizes, int n_in,
                              void* d_out, int out_size, void* d_ws, size_t ws_size,
                              hipStream_t stream) {
  const float* input   = (const float*)d_in[0];
  const float* other   = (const float*)d_in[1];
  const int*   rj      = (const int*)d_in[2];
  const int*   lj      = (const int*)d_in[3];
  const float* weights = (const float*)d_in[4];
  const float* Wi      = (const float*)d_in[5];
  const float* bi      = (const float*)d_in[6];
  const float* Wo      = (const float*)d_in[7];
  const float* We      = (const float*)d_in[8];
  const float* Wf      = (const float*)d_in[9];
  const float* bf      = (const float*)d_in[10];
  const float* Wout    = (const float*)d_in[11];
  const float* bout    = (const float*)d_in[12];

  const int D    = 64;
  const int N_IN = in_sizes[0] / D;
  const int N_OT = in_sizes[1] / D;
  const int E    = in_sizes[2];

  float* rhs    = (float*)d_ws;                       // [N_IN,64]
  float* lhs    = rhs + (size_t)N_IN * D;             // [N_OT,64]
  float* outAcc = lhs + (size_t)N_OT * D;             // [N_IN,64]

  const int tiles_in = (N_IN + 15) / 16;
  const int tiles_ot = (N_OT + 15) / 16;

  // rhs = input @ Wi + bi ; lhs = other @ Wo
  gemm64_kernel<<<(tiles_in + 7) / 8, 256, 0, stream>>>(input, Wi, bi, rhs, N_IN);
  gemm64_kernel<<<(tiles_ot + 7) / 8, 256, 0, stream>>>(other, Wo, nullptr, lhs, N_OT);

  // zero the segment accumulator
  const int n4 = (N_IN * D) / 4;
  zero4_kernel<<<(n4 + 255) / 256, 256, 0, stream>>>((float4*)outAcc, n4);

  // fused gather + edge-embed + leaky_relu + GEMM(Wf)+bf + segment scatter
  const int eTiles = (E + 15) / 16;
  edge_kernel<<<(eTiles + 7) / 8, 256, 0, stream>>>(rhs, lhs, rj, lj, weights,
                                                    We, Wf, bf, outAcc, E);

  // d_out = concat([outAcc, input]) @ Wout + bout
  gemm_concat128_kernel<<<(tiles_in + 7) / 8, 256, 0, stream>>>(
      outAcc, input, Wout, bout, (float*)d_out, N_IN);
}